// PhaseQuantizer_360777252996
// MI455X (gfx1250) — compile-verified
//
#include <hip/hip_runtime.h>

// ---------------------------------------------------------------------------
// PhaseQuantizer + checkerboard crosstalk stencil for MI455X (gfx1250)
//
// Memory-bound: 99 MB traffic -> ~4.3 us at 23.3 TB/s. The stencil collapses
// in linear site-index space to offsets {+-1, +-31, +-32} with validity a
// function of k = i % 63 only, once the LDS array has 32-wide zero guards.
// CDNA5 path: global_load_async_to_lds_b128 + s_wait_asynccnt staging.
// ---------------------------------------------------------------------------

#define M_SITES 2016      // 64*63/2 checkerboard sites per batch
#define THREADS 512       // 16 waves of 32
#define QUADS   504       // 2016 / 4 float4 quads per batch
#define GUARD   32        // zero guard on each side of the LDS site array
#define XTOT    (GUARD + M_SITES + GUARD)   // 2080 floats

constexpr float TWO_PI_F    = 6.28318530717958647692f;
constexpr float INV_2PI_F   = 0.15915494309189533577f;
constexpr float RATIO_F     = 6.28318530717958647692f / 255.0f;   // 2pi/(2^8-1)
constexpr float INV_RATIO_F = 255.0f / 6.28318530717958647692f;
constexpr double GAMMA_D    = 3.14159265358979323846 / (4.36 * 4.36);
constexpr float INV_GAMMA_F = (float)(1.0 / GAMMA_D);
constexpr float XTALK_F     = 0.01f;

// -------- CDNA5 async global->LDS path (probe via __has_builtin) -----------
#if defined(__has_builtin)
#if __has_builtin(__builtin_amdgcn_global_load_async_to_lds_b128)
#define USE_ASYNC_LDS 1
#endif
#endif
#ifndef USE_ASYNC_LDS
#define USE_ASYNC_LDS 0
#endif

#if USE_ASYNC_LDS
typedef int v4i __attribute__((ext_vector_type(4)));
typedef __attribute__((address_space(1))) v4i gv4i;   // global int4
typedef __attribute__((address_space(3))) v4i lv4i;   // LDS int4

__device__ __forceinline__ void async_cp_b128(const float* g, float* l) {
  __builtin_amdgcn_global_load_async_to_lds_b128(
      (gv4i*)g, (lv4i*)l, /*offset=*/0, /*cpol=*/0);
}
__device__ __forceinline__ void wait_async0() {
#if defined(__has_builtin) && __has_builtin(__builtin_amdgcn_s_wait_asynccnt)
  __builtin_amdgcn_s_wait_asynccnt(0);
#else
  asm volatile("s_wait_asynccnt 0" ::: "memory");
#endif
}
#endif

// -------- per-element quantize / gamma-scale math --------------------------
__device__ __forceinline__ float mod2pi(float x) {
  float q = floorf(x * INV_2PI_F);
  return fmaf(-q, TWO_PI_F, x);          // x - floor(x/2pi)*2pi
}

__device__ __forceinline__ float quant_scale(float p, float g) {
  float y = mod2pi(p);
  y = rintf(y * INV_RATIO_F) * RATIO_F;  // round-to-nearest-even, 8-bit grid
  y = y * (g * INV_GAMMA_F);             // * (noisy_gamma / GAMMA)
  return mod2pi(y);
}

// ---------------------------------------------------------------------------
__global__ __launch_bounds__(THREADS, 2)
void phase_xtalk_kernel(const float* __restrict__ phases,
                        const float* __restrict__ gammas,
                        float* __restrict__ out) {
  // [0,32) zero guard | [32, 32+2016) sites | [2048, 2080) zero guard
  __shared__ float X[XTOT];
#if USE_ASYNC_LDS
  __shared__ float rph[M_SITES];          // raw async-staged phases
#endif

  const size_t base = (size_t)blockIdx.x * M_SITES;
  const int t = threadIdx.x;

  // zero the guard zones (64 floats)
  if (t < GUARD)            X[t] = 0.0f;
  else if (t < 2 * GUARD)   X[GUARD + M_SITES + (t - GUARD)] = 0.0f;

  float xq[4];
  if (t < QUADS) {
    const int q4 = 4 * t;
    float4 ph, gm;
#if USE_ASYNC_LDS
    // CDNA5 async copy: each lane DMAs its 16B of phases straight into LDS.
    async_cp_b128(phases + base + q4, &rph[q4]);
    gm = *(const float4*)(gammas + base + q4);     // direct b128 to VGPRs
    wait_async0();
    ph = *(const float4*)&rph[q4];
#else
    ph = *(const float4*)(phases + base + q4);
    gm = *(const float4*)(gammas + base + q4);
#endif
    xq[0] = quant_scale(ph.x, gm.x);
    xq[1] = quant_scale(ph.y, gm.y);
    xq[2] = quant_scale(ph.z, gm.z);
    xq[3] = quant_scale(ph.w, gm.w);
    float4 xv = make_float4(xq[0], xq[1], xq[2], xq[3]);
    *(float4*)&X[GUARD + q4] = xv;        // ds_store_b128
  }

  __syncthreads();

  if (t < QUADS) {
    const int q4 = 4 * t;
    const int k0 = q4 - (q4 / 63) * 63;   // q4 % 63, one magic-div per thread
    float o[4];
#pragma unroll
    for (int j = 0; j < 4; ++j) {
      const int i = q4 + j;
      int k = k0 + j;
      if (k >= 63) k -= 63;
      // B points at X32[i-32]; all six neighbors at constant +ve offsets.
      const float* B = &X[i];             // X[i] == X32[i - 32]
      float nm32 = B[0];                  // i-32 : valid unless k==0
      float nm31 = B[1];                  // i-31 : valid unless k==31
      float nm1  = B[31];                 // i-1  : valid unless k==0 || k==32
      float np1  = B[33];                 // i+1  : valid unless k==31 || k==62
      float np31 = B[63];                 // i+31 : valid unless k==0
      float np32 = B[64];                 // i+32 : valid unless k==31
      const bool k00 = (k == 0), k31 = (k == 31);
      const bool km1 = k00 | (k == 32);   // kills i-1
      const bool kp1 = k31 | (k == 62);   // kills i+1
      float acc = (k00 ? 0.0f : nm32 + np31)
                + (k31 ? 0.0f : nm31 + np32)
                + (km1 ? 0.0f : nm1)
                + (kp1 ? 0.0f : np1);
      o[j] = fmaf(XTALK_F, acc, xq[j]);   // center tap weight 1.0, from regs
    }
    float4 ov = make_float4(o[0], o[1], o[2], o[3]);
    *(float4*)(out + base + q4) = ov;     // coalesced global_store_b128
  }
}

// ---------------------------------------------------------------------------
extern "C" void kernel_launch(void* const* d_in, const int* in_sizes, int n_in,
                              void* d_out, int out_size, void* d_ws, size_t ws_size,
                              hipStream_t stream) {
  const float* phases = (const float*)d_in[0];
  const float* gammas = (const float*)d_in[1];
  float* out = (float*)d_out;

  const int nbatch = in_sizes[0] / M_SITES;   // P*Q = 4096
  phase_xtalk_kernel<<<dim3(nbatch), dim3(THREADS), 0, stream>>>(phases, gammas, out);
}